// ParallelLoss_80745385165114
// MI455X (gfx1250) — compile-verified
//
#include <hip/hip_runtime.h>
#include <hip/hip_bf16.h>

#define BATCH 4
#define HH    320
#define WW    768
#define MAXD  3
#define TS    16
#define TD    (TS + 2 * MAXD)          /* 22 */
#define HW    (HH * WW)
#define NPIX  (BATCH * HH * WW)        /* 983040 */

typedef __attribute__((ext_vector_type(2))) float v2f;
typedef __attribute__((ext_vector_type(8))) float v8f;

__device__ __forceinline__ float luma255(const float* __restrict__ img, int ib, int idx) {
    // img + ib points at (b,0,0,0); channel stride HW
    return (0.299f * img[ib + idx] +
            0.587f * img[ib + HW + idx] +
            0.114f * img[ib + 2 * HW + idx]) * 255.0f;
}

__global__ __launch_bounds__(64) void zero_out_kernel(float* __restrict__ out) {
    if (threadIdx.x < 2) out[threadIdx.x] = 0.0f;
}

__global__ __launch_bounds__(256) void census_loss_kernel(
    const float* __restrict__ former,
    const float* __restrict__ latter,
    const float* __restrict__ flowF,
    const float* __restrict__ flowB,
    const float* __restrict__ fwM,
    const float* __restrict__ bwM,
    float* __restrict__ out)
{
    __shared__ float s1[TD * TD];   // gray(im1) tile+halo
    __shared__ float s2[TD * TD];   // gray(warp(im2)) tile+halo
    __shared__ float red[256];

    const int tid = threadIdx.x;
    const int zb  = blockIdx.z;
    const int b   = zb >> 1;
    const int dir = zb & 1;

    // dir 0: census(former) vs census(warp(latter, flowF)), fw_mask
    // dir 1: census(latter) vs census(warp(former, flowB)), bw_mask
    const float* im1  = dir ? latter : former;
    const float* im2  = dir ? former : latter;
    const float* flow = dir ? flowB  : flowF;
    const float* msk  = dir ? bwM    : fwM;

    const int ib = b * 3 * HW;   // image batch offset
    const int fb = b * 2 * HW;   // flow batch offset
    const int mb = b * HW;       // mask batch offset

    const int tx0 = blockIdx.x * TS - MAXD;
    const int ty0 = blockIdx.y * TS - MAXD;

    // ---- Stage halo: gray(im1) and bilinear-warped gray(im2) ----
    for (int i = tid; i < TD * TD; i += 256) {
        int ly = i / TD;
        int lx = i - ly * TD;
        int gy = min(max(ty0 + ly, 0), HH - 1);
        int gx = min(max(tx0 + lx, 0), WW - 1);
        int idx = gy * WW + gx;

        s1[i] = luma255(im1, ib, idx);

        float u  = flow[fb + idx];
        float v  = flow[fb + HW + idx];
        float fx = (float)gx + u;
        float fy = (float)gy + v;
        float xf = floorf(fx), yf = floorf(fy);
        float wx = fx - xf,   wy = fy - yf;
        int x0 = (int)fminf(fmaxf(xf,        0.0f), (float)(WW - 1));
        int x1 = (int)fminf(fmaxf(xf + 1.0f, 0.0f), (float)(WW - 1));
        int y0 = (int)fminf(fmaxf(yf,        0.0f), (float)(HH - 1));
        int y1 = (int)fminf(fmaxf(yf + 1.0f, 0.0f), (float)(HH - 1));

        float la = luma255(im2, ib, y0 * WW + x0);
        float lb = luma255(im2, ib, y0 * WW + x1);
        float lc = luma255(im2, ib, y1 * WW + x0);
        float ld = luma255(im2, ib, y1 * WW + x1);
        s2[i] = la * (1.0f - wx) * (1.0f - wy) + lb * wx * (1.0f - wy)
              + lc * (1.0f - wx) * wy          + ld * wx * wy;
    }
    __syncthreads();

    // ---- Per-pixel census distance + Charbonnier ----
    const int ty = tid >> 4, tx = tid & 15;
    const int py = blockIdx.y * TS + ty;
    const int px = blockIdx.x * TS + tx;
    const bool interior = (py >= MAXD) & (py < HH - MAXD) & (px >= MAXD) & (px < WW - MAXD);
    float m = interior ? msk[mb + py * WW + px] : 0.0f;

    float p = 0.0f;
    if (m != 0.0f) {
        const int ci = (ty + MAXD) * TD + (tx + MAXD);
        const float c1 = s1[ci];
        const float c2 = s2[ci];
        float dist = 0.0f;
        #pragma unroll
        for (int dy = -MAXD; dy <= MAXD; ++dy) {
            #pragma unroll
            for (int dx = -MAXD; dx <= MAXD; ++dx) {
                int ni = ci + dy * TD + dx;
                float a1 = s1[ni] - c1;
                float a2 = s2[ni] - c2;
                float t1 = a1 * __builtin_amdgcn_rsqf(0.81f + a1 * a1);
                float t2 = a2 * __builtin_amdgcn_rsqf(0.81f + a2 * a2);
                float df = t1 - t2;
                float d  = df * df;
                dist += d * __builtin_amdgcn_rcpf(0.1f + d);
            }
        }
        // Charbonnier: (dist^2 + 0.001^2)^0.45 * mask
        p = __powf(dist * dist + 1.0e-6f, 0.45f) * m;
    }

    red[tid] = p;
    __syncthreads();

    // ---- Block reduction: LDS 256->32, WMMA f32 ones x data, shfl finish ----
    if (tid < 32) {                 // whole wave 0: EXEC all-ones for WMMA
        float v = 0.0f;
        #pragma unroll
        for (int w = 0; w < 8; ++w) v += red[tid + 32 * w];

        v2f a;  a[0] = 1.0f; a[1] = 1.0f;   // A = ones(16x4)
        v2f bb; bb[0] = v;   bb[1] = 0.0f;  // B holds each lane partial exactly once
        v8f c = {};
        // D[m][n] = sum_k B[k][n]  (column sums of B, replicated over rows)
        c = __builtin_amdgcn_wmma_f32_16x16x4_f32(
                /*neg_a=*/false, a, /*neg_b=*/false, bb,
                /*c_mod=*/(short)0, c, /*reuse_a=*/false, /*reuse_b=*/false);

        float t = c[0];             // lane n (and n+16): colsum(n)
        t += __shfl_xor(t, 1, 32);
        t += __shfl_xor(t, 2, 32);
        t += __shfl_xor(t, 4, 32);
        t += __shfl_xor(t, 8, 32);  // lane 0: sum over columns 0..15 = all 32 partials

        if (tid == 0)
            atomicAdd(&out[dir], t * (1.0f / (float)NPIX));
    }
}

extern "C" void kernel_launch(void* const* d_in, const int* in_sizes, int n_in,
                              void* d_out, int out_size, void* d_ws, size_t ws_size,
                              hipStream_t stream) {
    (void)in_sizes; (void)n_in; (void)out_size; (void)d_ws; (void)ws_size;
    const float* former = (const float*)d_in[0];
    const float* latter = (const float*)d_in[1];
    const float* flowF  = (const float*)d_in[2];
    const float* flowB  = (const float*)d_in[3];
    const float* fwM    = (const float*)d_in[4];
    const float* bwM    = (const float*)d_in[5];
    float* out = (float*)d_out;

    zero_out_kernel<<<1, 64, 0, stream>>>(out);

    dim3 grid(WW / TS, HH / TS, BATCH * 2);   // 48 x 20 x 8
    census_loss_kernel<<<grid, 256, 0, stream>>>(
        former, latter, flowF, flowB, fwM, bwM, out);
}